// QModel_15745350107852
// MI455X (gfx1250) — compile-verified
//
#include <hip/hip_runtime.h>

// Analytic collapse of the quantum circuit (N=10 qubits, B=16384):
//   - Only RX on qubits 0,1 creates amplitude; the state is supported on
//     exactly 4 basis states with p(b0,b1) = Bernoulli(sin^2(xa/2)) x
//     Bernoulli(sin^2(ya/2)).
//   - CNOTs are basis permutations, RZs are diagonal unit-modulus phases,
//     so probabilities are exactly independent of theta.
//   - Three applications of the CNOT chain = prefix-XOR^3 on bit strings,
//     mapping initial bits (b0,b1,0,...) to measured bits:
//       q0 = b0, q1 = b0^b1, q2 = b1.
//   => <Z0> = cos(xa), <Z1> = cos(xa)cos(ya), <Z2> = cos(ya),
//      with xa = 2*pi*x0 - pi  =>  cos(xa) = -cos(2*pi*x0)  (same for ya).
//
// out[b,0] = (1 - cos(2*pi*x0))/2
// out[b,1] = (1 + cos(2*pi*x0)*cos(2*pi*x1))/2
// out[b,2] = (1 - cos(2*pi*x1))/2
//
// Roofline: 320 KB total traffic ~ 14 ns at 23.3 TB/s HBM -> kernel is
// launch-latency bound. No GEMM exists in the optimal formulation, so the
// WMMA units are intentionally not used (any v_wmma here would be dead work).

struct Out3 { float a, b, c; };   // 12-byte row, 4-byte aligned

__global__ __launch_bounds__(256) void qcircuit_analytic(
    const float2* __restrict__ x,   // (B,2) row-major -> one float2 per row
    Out3* __restrict__ out,         // (B,3) row-major
    int B)
{
    int b = blockIdx.x * blockDim.x + threadIdx.x;
    if (b >= B) return;

    float2 xv = x[b];
    // cos(2*pi*x), accurately range-reduced; the two calls vectorize into
    // packed f32 ops on gfx1250.
    float c0 = cospif(2.0f * xv.x);
    float c1 = cospif(2.0f * xv.y);

    Out3 o;
    o.a = 0.5f - 0.5f * c0;
    o.b = fmaf(0.5f * c0, c1, 0.5f);
    o.c = 0.5f - 0.5f * c1;
    out[b] = o;   // single 12-byte aggregate store -> b96 / b64+b32
}

extern "C" void kernel_launch(void* const* d_in, const int* in_sizes, int n_in,
                              void* d_out, int out_size, void* d_ws, size_t ws_size,
                              hipStream_t stream)
{
    (void)n_in; (void)d_ws; (void)ws_size; (void)out_size;

    const float2* x = (const float2*)d_in[0];   // (B,2) float32
    // d_in[1] = theta (30 floats): provably unused -- RZ phases cancel in
    // |amplitude|^2 and CNOT permutations are theta-independent.
    Out3* out = (Out3*)d_out;                   // (B,3) float32

    int B = in_sizes[0] / 2;
    int threads = 256;
    int blocks = (B + threads - 1) / threads;
    qcircuit_analytic<<<blocks, threads, 0, stream>>>(x, out, B);
}